// GVAE_Encoder_33294586479244
// MI455X (gfx1250) — compile-verified
//
#include <hip/hip_runtime.h>
#include <math.h>

typedef __attribute__((ext_vector_type(2))) float v2f;
typedef __attribute__((ext_vector_type(8))) float v8f;

// ---------------- degree / symmetric norm ----------------
__global__ void k_deg_init(float* deg, int n) {
  int i = blockIdx.x * blockDim.x + threadIdx.x;
  if (i < n) deg[i] = 1.0f;  // self loop contributes 1
}

__global__ void k_deg_count(const int* __restrict__ dst, float* deg, int e) {
  int i = blockIdx.x * blockDim.x + threadIdx.x;
  if (i < e) unsafeAtomicAdd(&deg[dst[i]], 1.0f);
}

__global__ void k_dinv(float* deg, int n) {
  int i = blockIdx.x * blockDim.x + threadIdx.x;
  if (i < n) deg[i] = rsqrtf(fmaxf(deg[i], 1.0f));
}

// ---------------- WMMA f32 GEMM with LDS-staged B ----------------
// grid.x covers row tiles (one wave = 16 rows), grid.y covers 64-col strips.
// All 8 waves of a block share one [KC x 64] B tile in LDS per K-chunk.
// M multiple of 16 (50000 = 16*3125); K multiple of KC (512, 128).
#define GEMM_KC 128
__global__ void __launch_bounds__(256) k_gemm_wmma(
    const float* __restrict__ A, const float* __restrict__ B,
    float* __restrict__ C, int M, int K, int ldb, int ldc) {
  __shared__ float Bs[GEMM_KC * 64];  // 32 KB

  int lane = (int)(threadIdx.x & 31);
  int rowTile = blockIdx.x * (int)(blockDim.x >> 5) + (int)(threadIdx.x >> 5);
  int n0 = (int)blockIdx.y << 6;
  int numRowTiles = M >> 4;
  bool valid = rowTile < numRowTiles;   // wave-uniform

  int m0 = (valid ? rowTile : 0) << 4;
  int nl = lane & 15;                   // row (A) / col (B,C) within 16
  int kh = (lane >> 4) << 1;            // 0 for lanes 0-15, 2 for lanes 16-31
  const float* Arow = A + (size_t)(m0 + nl) * K + kh;

  v8f acc0 = {}, acc1 = {}, acc2 = {}, acc3 = {};

  for (int kc = 0; kc < K; kc += GEMM_KC) {
    // cooperative stage: Bs[kk][c] = B[kc+kk][n0+c]
    for (int i = (int)threadIdx.x; i < GEMM_KC * 64; i += (int)blockDim.x) {
      int kk = i >> 6;
      int c  = i & 63;
      Bs[i] = B[(size_t)(kc + kk) * ldb + n0 + c];
    }
    __syncthreads();

    if (valid) {
      for (int k0 = 0; k0 < GEMM_KC; k0 += 4) {
        v2f a = *(const v2f*)(Arow + kc + k0);        // (k+kh, k+kh+1) of this lane's row
        const float* L0 = Bs + (k0 + kh) * 64 + nl;   // row k+kh
        const float* L1 = L0 + 64;                    // row k+kh+1
        v2f b0 = { L0[0],  L1[0]  };
        v2f b1 = { L0[16], L1[16] };
        v2f b2 = { L0[32], L1[32] };
        v2f b3 = { L0[48], L1[48] };
        acc0 = __builtin_amdgcn_wmma_f32_16x16x4_f32(false, a, false, b0, (short)0, acc0, false, false);
        acc1 = __builtin_amdgcn_wmma_f32_16x16x4_f32(false, a, false, b1, (short)0, acc1, false, false);
        acc2 = __builtin_amdgcn_wmma_f32_16x16x4_f32(false, a, false, b2, (short)0, acc2, false, false);
        acc3 = __builtin_amdgcn_wmma_f32_16x16x4_f32(false, a, false, b3, (short)0, acc3, false, false);
      }
    }
    __syncthreads();
  }

  if (!valid) return;
  int rbase = m0 + ((lane >> 4) << 3);  // lanes 16-31 hold rows m0+8..m0+15
  float* Cp = C + (size_t)rbase * ldc + n0 + nl;
#pragma unroll
  for (int r = 0; r < 8; ++r) {
    float* row = Cp + (size_t)r * ldc;
    row[0]  = acc0[r];
    row[16] = acc1[r];
    row[32] = acc2[r];
    row[48] = acc3[r];
  }
}

// ---------------- propagation ----------------
// out[n,c] = in[n,c]*dinv[n]^2 + bias[c]   (self-loop term + post-sum bias, F=128)
__global__ void k_prop_init(const float* __restrict__ in, float* __restrict__ out,
                            const float* __restrict__ dinv, const float* __restrict__ bias,
                            int n) {
  int gid = blockIdx.x * blockDim.x + threadIdx.x;
  if (gid >= n * 128) return;
  int node = gid >> 7;
  int c = gid & 127;
  float di = dinv[node];
  out[gid] = in[gid] * di * di + bias[c];
}

// one wave per edge: out[dst] += in[src] * dinv[src]*dinv[dst]  (F=128)
__global__ void k_prop_edges(const float* __restrict__ in, float* __restrict__ out,
                             const int* __restrict__ src, const int* __restrict__ dst,
                             const float* __restrict__ dinv, int e) {
  int gid = blockIdx.x * blockDim.x + threadIdx.x;
  int eid = gid >> 5;
  int lane = gid & 31;
  if (eid >= e) return;
  int s = src[eid];
  int d = dst[eid];
  float w = dinv[s] * dinv[d];
  const float* pin = in + (size_t)s * 128;
  float* pout = out + (size_t)d * 128;
#pragma unroll
  for (int j = 0; j < 4; ++j) {
    int c = lane + j * 32;
    unsafeAtomicAdd(&pout[c], pin[c] * w);
  }
}

// ---------------- misc ----------------
__global__ void k_bcat(const float* __restrict__ b2, const float* __restrict__ b3,
                       float* __restrict__ bcat) {
  int t = threadIdx.x;  // launched with 128 threads
  bcat[t] = (t < 64) ? b2[t] : b3[t - 64];
}

// z[n,c] = eps[n,c]*exp(logvar[n,c]) + mu[n,c]; agg is [N,128]: cols 0-63=mu, 64-127=logvar
__global__ void k_final(const float* __restrict__ agg, const float* __restrict__ eps,
                        float* __restrict__ z, int n) {
  int gid = blockIdx.x * blockDim.x + threadIdx.x;
  if (gid >= n * 64) return;
  int node = gid >> 6;
  int c = gid & 63;
  float mu = agg[(size_t)node * 128 + c];
  float lv = agg[(size_t)node * 128 + 64 + c];
  z[gid] = eps[gid] * expf(lv) + mu;
}

static inline int cdiv_i(long long a, long long b) { return (int)((a + b - 1) / b); }

extern "C" void kernel_launch(void* const* d_in, const int* in_sizes, int n_in,
                              void* d_out, int out_size, void* d_ws, size_t ws_size,
                              hipStream_t stream) {
  const float* x   = (const float*)d_in[0];   // [N, 512]
  const int*   ei  = (const int*)d_in[1];     // [2, E]
  const float* W1  = (const float*)d_in[2];   // [512, 128]
  const float* b1  = (const float*)d_in[3];   // [128]
  const float* W2  = (const float*)d_in[4];   // [128, 64]
  const float* b2  = (const float*)d_in[5];   // [64]
  const float* W3  = (const float*)d_in[6];   // [128, 64]
  const float* b3  = (const float*)d_in[7];   // [64]
  const float* eps = (const float*)d_in[8];   // [N, 64]
  float* z = (float*)d_out;

  const int Nn = in_sizes[0] / 512;           // 50000 (multiple of 16)
  const int E  = in_sizes[1] / 2;             // 1.6M
  const int* src = ei;
  const int* dst = ei + E;

  // workspace layout (floats)
  float* ws = (float*)d_ws;
  size_t o = 0;
  float* dinv = ws + o;  o += ((size_t)Nn + 255) & ~(size_t)255;
  float* hw1  = ws + o;  o += (size_t)Nn * 128;   // x@W1
  float* agg1 = ws + o;  o += (size_t)Nn * 128;   // h1
  float* hw23 = ws + o;  o += (size_t)Nn * 128;   // [h1@W2 | h1@W3]
  float* agg23= ws + o;  o += (size_t)Nn * 128;   // [mu | logvar]
  float* bcat = ws + o;  o += 128;

  const int T = 256;
  const int rowTiles = Nn >> 4;               // 3125
  const int gemmBlocksX = cdiv_i(rowTiles, T / 32);

  // 1) symmetric normalization dinv = rsqrt(deg)
  k_deg_init <<<cdiv_i(Nn, T), T, 0, stream>>>(dinv, Nn);
  k_deg_count<<<cdiv_i(E,  T), T, 0, stream>>>(dst, dinv, E);
  k_dinv     <<<cdiv_i(Nn, T), T, 0, stream>>>(dinv, Nn);

  // 2) hw1 = x @ W1   (M=Nn, K=512, 128 cols -> grid.y = 2 strips)
  k_gemm_wmma<<<dim3(gemmBlocksX, 2), T, 0, stream>>>(x, W1, hw1, Nn, 512, 128, 128);

  // 3) h1 = propagate(hw1) + b1
  k_prop_init <<<cdiv_i((long long)Nn * 128, T), T, 0, stream>>>(hw1, agg1, dinv, b1, Nn);
  k_prop_edges<<<cdiv_i((long long)E * 32, T),  T, 0, stream>>>(hw1, agg1, src, dst, dinv, E);

  // 4) hw23[:,0:64] = h1 @ W2 ; hw23[:,64:128] = h1 @ W3
  k_gemm_wmma<<<dim3(gemmBlocksX, 1), T, 0, stream>>>(agg1, W2, hw23,      Nn, 128, 64, 128);
  k_gemm_wmma<<<dim3(gemmBlocksX, 1), T, 0, stream>>>(agg1, W3, hw23 + 64, Nn, 128, 64, 128);

  // 5) [mu | logvar] = propagate(hw23) + [b2|b3]
  k_bcat<<<1, 128, 0, stream>>>(b2, b3, bcat);
  k_prop_init <<<cdiv_i((long long)Nn * 128, T), T, 0, stream>>>(hw23, agg23, dinv, bcat, Nn);
  k_prop_edges<<<cdiv_i((long long)E * 32, T),  T, 0, stream>>>(hw23, agg23, src, dst, dinv, E);

  // 6) z = eps*exp(logvar) + mu
  k_final<<<cdiv_i((long long)Nn * 64, T), T, 0, stream>>>(agg23, eps, z, Nn);
}